// LGCNEncoder_77790447665861
// MI455X (gfx1250) — compile-verified
//
#include <hip/hip_runtime.h>
#include <stdint.h>

// LightGCN propagation on MI455X (gfx1250).
// Memory-bound SpMM with L2-resident (38.4MB << 192MB L2) gather/scatter.
// CDNA5 paths used: async global->LDS edge staging (ASYNCcnt double buffer),
// wave32 row mapping (32 lanes x float2 = one 64-float row), saddr-form
// scalar addressing via readfirstlane, global_prefetch_b8, and agent-scope
// non-returning f32 atomics (resolved in L2).
// WMMA is deliberately NOT used: there is no dense GEMM in this op; the
// bottleneck is L2 atomic throughput, and a diag(v)xG WMMA formulation would
// waste 32x the FLOPs plus layout shuffles for zero roofline benefit.

static constexpr int kUserNum = 100000;
static constexpr int kItemNum = 50000;
static constexpr int kNodes   = kUserNum + kItemNum;
static constexpr int kEmb     = 64;
static constexpr int kLayers  = 3;

static constexpr int TPB   = 256;        // 8 waves/block (wave32)
static constexpr int WPB   = TPB / 32;
static constexpr int CHUNK = 128;        // edges staged per async step per wave

// ---------------- SpMM: gather -> scale -> atomic scatter-add ----------------

// Stage 512B (32 lanes x b128) of an edge array into LDS asynchronously.
// GVS addressing: mem = SADDR(base) + VADDR(per-lane byte offset); LDS dest is
// the per-lane LDS byte address in the VDST VGPR. Tracked by ASYNCcnt.
__device__ __forceinline__ void async_stage(const void* gbase, uint32_t byteOff,
                                            uint32_t ldsAddr) {
  asm volatile("global_load_async_to_lds_b128 %0, %1, %2"
               :: "v"(ldsAddr), "v"(byteOff), "s"(gbase)
               : "memory");
}

__global__ __launch_bounds__(TPB)
void lgcn_spmm(const float* __restrict__ cur, float* __restrict__ nxt,
               const float* __restrict__ vals, const int* __restrict__ src,
               const int* __restrict__ dst, int nEdges) {
  // Double-buffered per-wave staging buffers: 3 arrays x 2 bufs x 128 edges.
  __shared__ uint32_t lsrc[WPB][2][CHUNK];
  __shared__ uint32_t ldst[WPB][2][CHUNK];
  __shared__ uint32_t lval[WPB][2][CHUNK];

  const int  lane = threadIdx.x & 31;
  const int  wav  = threadIdx.x >> 5;
  const long gw   = (long)blockIdx.x * WPB + wav;
  const long nw   = (long)gridDim.x * WPB;
  const long stride = nw * CHUNK;

  // Per-lane LDS byte addresses for the b128 async destinations.
  uint32_t aS[2], aD[2], aV[2];
#pragma unroll
  for (int b = 0; b < 2; ++b) {
    aS[b] = (uint32_t)(uintptr_t)&lsrc[wav][b][lane * 4];
    aD[b] = (uint32_t)(uintptr_t)&ldst[wav][b][lane * 4];
    aV[b] = (uint32_t)(uintptr_t)&lval[wav][b][lane * 4];
  }

  long e0  = gw * CHUNK;
  bool have = (e0 + CHUNK) <= (long)nEdges;
  if (have) {
    uint32_t off = (uint32_t)((e0 + (long)lane * 4) * 4);
    async_stage(src,  off, aS[0]);
    async_stage(dst,  off, aD[0]);
    async_stage(vals, off, aV[0]);
  }
  int pb = 0;
  while (have) {
    long e1 = e0 + stride;
    bool haveNext = (e1 + CHUNK) <= (long)nEdges;
    if (haveNext) {
      // Issue next chunk into the other buffer, then wait only for the
      // current chunk (in-order completion => <=3 outstanding means chunk
      // `pb` has fully landed in LDS).
      uint32_t off = (uint32_t)((e1 + (long)lane * 4) * 4);
      async_stage(src,  off, aS[pb ^ 1]);
      async_stage(dst,  off, aD[pb ^ 1]);
      async_stage(vals, off, aV[pb ^ 1]);
      asm volatile("s_wait_asynccnt 3" ::: "memory");
    } else {
      asm volatile("s_wait_asynccnt 0" ::: "memory");
    }

    // Raise MLP: each lane prefetches the head of 4 upcoming gather rows.
#pragma unroll
    for (int q = 0; q < 4; ++q) {
      uint32_t srow = lsrc[wav][pb][lane + q * 32];
      __builtin_prefetch(cur + (size_t)srow * kEmb, 0, 3);
    }

    // One edge per inner step: whole wave handles the 64-float row
    // (lane -> float2). Uniform LDS reads + readfirstlane put the row
    // index/value into SGPRs so loads/atomics use saddr-form addressing.
#pragma unroll 4
    for (int j = 0; j < CHUNK; ++j) {
      uint32_t s = (uint32_t)__builtin_amdgcn_readfirstlane((int)lsrc[wav][pb][j]);
      uint32_t d = (uint32_t)__builtin_amdgcn_readfirstlane((int)ldst[wav][pb][j]);
      float    v = __uint_as_float(
                     (uint32_t)__builtin_amdgcn_readfirstlane((int)lval[wav][pb][j]));
      const float2 x = *(const float2*)(cur + (size_t)s * kEmb + lane * 2);
      float* p = nxt + (size_t)d * kEmb + lane * 2;
      __hip_atomic_fetch_add(p,     v * x.x, __ATOMIC_RELAXED, __HIP_MEMORY_SCOPE_AGENT);
      __hip_atomic_fetch_add(p + 1, v * x.y, __ATOMIC_RELAXED, __HIP_MEMORY_SCOPE_AGENT);
    }
    e0 = e1; pb ^= 1; have = haveNext;
  }

  // Generic tail (nEdges % CHUNK != 0) — dead for E = 4,000,000 but kept for
  // correctness under arbitrary sizes. One wave walks the remainder directly.
  long tailStart = ((long)nEdges / CHUNK) * CHUNK;
  int  tailLen   = (int)((long)nEdges - tailStart);
  if (tailLen && ((tailStart / CHUNK) % nw) == gw) {
    for (int j = 0; j < tailLen; ++j) {
      int   s = src[tailStart + j];
      int   d = dst[tailStart + j];
      float v = vals[tailStart + j];
      const float2 x = *(const float2*)(cur + (size_t)s * kEmb + lane * 2);
      float* p = nxt + (size_t)d * kEmb + lane * 2;
      __hip_atomic_fetch_add(p,     v * x.x, __ATOMIC_RELAXED, __HIP_MEMORY_SCOPE_AGENT);
      __hip_atomic_fetch_add(p + 1, v * x.y, __ATOMIC_RELAXED, __HIP_MEMORY_SCOPE_AGENT);
    }
  }
}

// ---------------- elementwise kernels (float4 = b128 traffic) ----------------

__global__ __launch_bounds__(TPB)
void lgcn_init(const float4* __restrict__ u4, const float4* __restrict__ i4,
               float4* __restrict__ cur4, float4* __restrict__ acc4,
               int total4, int user4) {
  int i = blockIdx.x * TPB + threadIdx.x;
  if (i >= total4) return;
  float4 v;
  if (i < user4) v = u4[i];
  else           v = i4[i - user4];
  cur4[i] = v;   // layer-0 embedding
  acc4[i] = v;   // running sum starts at emb^0
}

__global__ __launch_bounds__(TPB)
void lgcn_zero(float4* __restrict__ p, int total4) {
  int i = blockIdx.x * TPB + threadIdx.x;
  if (i < total4) p[i] = make_float4(0.f, 0.f, 0.f, 0.f);
}

__global__ __launch_bounds__(TPB)
void lgcn_add_scale(float4* __restrict__ acc, const float4* __restrict__ nxt,
                    float s, int total4) {
  int i = blockIdx.x * TPB + threadIdx.x;
  if (i >= total4) return;
  float4 a = acc[i], b = nxt[i];
  a.x = (a.x + b.x) * s;
  a.y = (a.y + b.y) * s;
  a.z = (a.z + b.z) * s;
  a.w = (a.w + b.w) * s;
  acc[i] = a;
}

// ---------------- host-side orchestration ----------------

extern "C" void kernel_launch(void* const* d_in, const int* in_sizes, int n_in,
                              void* d_out, int out_size, void* d_ws, size_t ws_size,
                              hipStream_t stream) {
  const float* user_emb  = (const float*)d_in[0];
  const float* item_emb  = (const float*)d_in[1];
  const float* edge_vals = (const float*)d_in[2];
  const int*   edge_src  = (const int*)d_in[3];
  const int*   edge_dst  = (const int*)d_in[4];
  const int    nEdges    = in_sizes[2];

  float* out  = (float*)d_out;                       // acc / final_emb [N, 64]
  float* bufA = (float*)d_ws;                        // cur  [N, 64]
  float* bufB = bufA + (size_t)kNodes * kEmb;        // next [N, 64] (ws >= 76.8MB)

  const int total4 = kNodes * kEmb / 4;              // 2,400,000 float4
  const int user4  = kUserNum * kEmb / 4;
  const int ewGrid = (total4 + TPB - 1) / TPB;

  lgcn_init<<<ewGrid, TPB, 0, stream>>>(
      (const float4*)user_emb, (const float4*)item_emb,
      (float4*)bufA, (float4*)out, total4, user4);

  float* cur = bufA;
  float* nxt = bufB;
  const int spmmGrid = 1024;                         // 8192 waves in flight
  for (int l = 0; l < kLayers; ++l) {
    lgcn_zero<<<ewGrid, TPB, 0, stream>>>((float4*)nxt, total4);
    lgcn_spmm<<<spmmGrid, TPB, 0, stream>>>(cur, nxt, edge_vals,
                                            edge_src, edge_dst, nEdges);
    const float s = (l == kLayers - 1) ? (1.0f / (kLayers + 1)) : 1.0f;
    lgcn_add_scale<<<ewGrid, TPB, 0, stream>>>((float4*)out, (const float4*)nxt,
                                               s, total4);
    float* t = cur; cur = nxt; nxt = t;
  }
  (void)n_in; (void)out_size; (void)ws_size;
}